// GCNEncoder_12232066859181
// MI455X (gfx1250) — compile-verified
//
#include <hip/hip_runtime.h>
#include <hip/hip_bf16.h>

typedef float v2f __attribute__((ext_vector_type(2)));
typedef float v8f __attribute__((ext_vector_type(8)));

// ---------------------------------------------------------------------------
// GEMM1: XW1[N,64] = X[N,128] @ W1[128,64], fp32 WMMA 16x16x4.
// Block = 128 threads = 4 waves; wave w computes the 16x16 tile at
// rows blockIdx.x*16, cols w*16. K loop: 32 x V_WMMA_F32_16X16X4_F32.
// ---------------------------------------------------------------------------
__global__ void __launch_bounds__(128)
gcn_gemm_xw1(const float* __restrict__ X, const float* __restrict__ W1,
             float* __restrict__ XW1, int N) {
    const int lane = threadIdx.x & 31;
    const int wave = threadIdx.x >> 5;          // 0..3 -> col tile
    const int row0 = blockIdx.x * 16;
    const int col0 = wave * 16;
    const int r    = lane & 15;                  // M for A, N for B/C
    const int kh   = lane >> 4;                  // K half-select (0/1)

    const int rowA = min(row0 + r, N - 1);       // clamped load row (tail safe)

    v8f acc = {};
#pragma unroll
    for (int k = 0; k < 128; k += 4) {
        // A: lane holds X[rowA, k + 2*kh + {0,1}]  (contiguous pair)
        v2f a = *(const v2f*)(X + (size_t)rowA * 128 + k + 2 * kh);
        // B: lane holds W1[k + 2*kh + {0,1}, col0 + r]  (stride-64 pair)
        v2f b;
        b.x = W1[(size_t)(k + 2 * kh + 0) * 64 + col0 + r];
        b.y = W1[(size_t)(k + 2 * kh + 1) * 64 + col0 + r];
        acc = __builtin_amdgcn_wmma_f32_16x16x4_f32(
            false, a, false, b, (short)0, acc, false, false);
    }

    // D layout: VGPR v -> row (v + 8*kh), col r
    const int mbase = kh * 8;
#pragma unroll
    for (int v = 0; v < 8; ++v) {
        int row = row0 + mbase + v;
        if (row < N) XW1[(size_t)row * 64 + col0 + r] = acc[v];
    }
}

// ---------------------------------------------------------------------------
// GEMM2: H2in[N,32] = relu(H1pre[N,64]) @ W2[64,32], fp32 WMMA, ReLU fused
// into the A-operand load. Block = 64 threads = 2 waves = 2 col tiles.
// ---------------------------------------------------------------------------
__global__ void __launch_bounds__(64)
gcn_gemm_h1w2(const float* __restrict__ H1pre, const float* __restrict__ W2,
              float* __restrict__ H2, int N) {
    const int lane = threadIdx.x & 31;
    const int wave = threadIdx.x >> 5;          // 0..1 -> col tile
    const int row0 = blockIdx.x * 16;
    const int col0 = wave * 16;
    const int r    = lane & 15;
    const int kh   = lane >> 4;

    const int rowA = min(row0 + r, N - 1);

    v8f acc = {};
#pragma unroll
    for (int k = 0; k < 64; k += 4) {
        v2f a = *(const v2f*)(H1pre + (size_t)rowA * 64 + k + 2 * kh);
        a.x = fmaxf(a.x, 0.0f);                  // fused ReLU
        a.y = fmaxf(a.y, 0.0f);
        v2f b;
        b.x = W2[(size_t)(k + 2 * kh + 0) * 32 + col0 + r];
        b.y = W2[(size_t)(k + 2 * kh + 1) * 32 + col0 + r];
        acc = __builtin_amdgcn_wmma_f32_16x16x4_f32(
            false, a, false, b, (short)0, acc, false, false);
    }

    const int mbase = kh * 8;
#pragma unroll
    for (int v = 0; v < 8; ++v) {
        int row = row0 + mbase + v;
        if (row < N) H2[(size_t)row * 32 + col0 + r] = acc[v];
    }
}

// ---------------------------------------------------------------------------
// SpMM, D=64: one wave32 per edge, each lane owns a float2 of the feature
// row. out[dst] += w * in[src] via hardware global_atomic_add_f32.
// src/dst/w are wave-uniform -> compiler scalarizes the loads.
// ---------------------------------------------------------------------------
__global__ void __launch_bounds__(256)
gcn_spmm64(const int* __restrict__ src, const int* __restrict__ dst,
           const float* __restrict__ ew, const float* __restrict__ Xin,
           float* __restrict__ Yout, int E) {
    const unsigned gid  = blockIdx.x * blockDim.x + threadIdx.x;
    const int      e    = (int)(gid >> 5);
    const int      lane = (int)(gid & 31);
    if (e >= E) return;
    const int   s = src[e];
    const int   d = dst[e];
    const float w = ew[e];
    v2f v = *(const v2f*)(Xin + (size_t)s * 64 + lane * 2);
    float* p = Yout + (size_t)d * 64 + lane * 2;
    unsafeAtomicAdd(p + 0, w * v.x);
    unsafeAtomicAdd(p + 1, w * v.y);
}

// SpMM, D=32: one wave32 per edge, one float per lane.
__global__ void __launch_bounds__(256)
gcn_spmm32(const int* __restrict__ src, const int* __restrict__ dst,
           const float* __restrict__ ew, const float* __restrict__ Xin,
           float* __restrict__ Yout, int E) {
    const unsigned gid  = blockIdx.x * blockDim.x + threadIdx.x;
    const int      e    = (int)(gid >> 5);
    const int      lane = (int)(gid & 31);
    if (e >= E) return;
    const int   s = src[e];
    const int   d = dst[e];
    const float w = ew[e];
    float v = Xin[(size_t)s * 32 + lane];
    unsafeAtomicAdd(Yout + (size_t)d * 32 + lane, w * v);
}

// ---------------------------------------------------------------------------
// Orchestration. Inputs (setup_inputs order):
//   d_in[0] features [N,128] f32    d_in[1] edge_weight [E] f32
//   d_in[2] W1 [128,64] f32         d_in[3] W2 [64,32] f32
//   d_in[4] src [E] i32             d_in[5] dst [E] i32
// d_out: embeddings [N,32] f32.
// ---------------------------------------------------------------------------
extern "C" void kernel_launch(void* const* d_in, const int* in_sizes, int n_in,
                              void* d_out, int out_size, void* d_ws, size_t ws_size,
                              hipStream_t stream) {
    const float* X  = (const float*)d_in[0];
    const float* ew = (const float*)d_in[1];
    const float* W1 = (const float*)d_in[2];
    const float* W2 = (const float*)d_in[3];
    const int*   sr = (const int*)d_in[4];
    const int*   ds = (const int*)d_in[5];
    float*       out = (float*)d_out;

    const int N = in_sizes[0] / 128;   // 100000
    const int E = in_sizes[1];         // 1600000

    // Workspace carve-up: xw1[N*64] | h1pre[N*64] | h2in[N*32]
    float* xw1   = (float*)d_ws;
    float* h1pre = xw1 + (size_t)N * 64;
    float* h2in  = h1pre + (size_t)N * 64;

    // Zero the segment-sum accumulators (capture-safe async memsets).
    hipMemsetAsync(h1pre, 0, (size_t)N * 64 * sizeof(float), stream);
    hipMemsetAsync(out,   0, (size_t)N * 32 * sizeof(float), stream);

    const int rowTiles = (N + 15) / 16;

    // 1) XW1 = X @ W1           (WMMA f32)
    gcn_gemm_xw1<<<rowTiles, 128, 0, stream>>>(X, W1, xw1, N);

    // 2) h1pre = segment_sum over edges of w * XW1[src]   (atomics)
    {
        unsigned threads = (unsigned)E * 32u;
        unsigned blocks  = (threads + 255u) / 256u;
        gcn_spmm64<<<blocks, 256, 0, stream>>>(sr, ds, ew, xw1, h1pre, E);
    }

    // 3) h2in = relu(h1pre) @ W2   (WMMA f32, fused ReLU)
    gcn_gemm_h1w2<<<rowTiles, 64, 0, stream>>>(h1pre, W2, h2in, N);

    // 4) out = segment_sum over edges of w * h2in[src]    (atomics)
    {
        unsigned threads = (unsigned)E * 32u;
        unsigned blocks  = (threads + 255u) / 256u;
        gcn_spmm32<<<blocks, 256, 0, stream>>>(sr, ds, ew, h2in, out, E);
    }
}